// GCNEncoder_67336497266937
// MI455X (gfx1250) — compile-verified
//
#include <hip/hip_runtime.h>
#include <hip/hip_bf16.h>

typedef float v2f __attribute__((ext_vector_type(2)));
typedef float v8f __attribute__((ext_vector_type(8)));

#define FD 64            // feature dim (fixed by reference)
#define WPB 8            // waves per block (256 threads, wave32)

// ---------------------------------------------------------------- degree ----
__global__ void __launch_bounds__(256) deg_init_kernel(unsigned* __restrict__ deg, int N) {
  int i = blockIdx.x * 256 + threadIdx.x;
  if (i < N) deg[i] = 1u;                       // self-loop contributes 1
}

__global__ void __launch_bounds__(256) deg_count_kernel(const int* __restrict__ ei, int E,
                                                        unsigned* __restrict__ deg) {
  int e = blockIdx.x * 256 + threadIdx.x;
  if (e < E) atomicAdd(&deg[ei[E + e]], 1u);    // dst row of edge_index
}

__global__ void __launch_bounds__(256) dinv_kernel(const unsigned* __restrict__ deg,
                                                   float* __restrict__ dinv, int N) {
  int i = blockIdx.x * 256 + threadIdx.x;
  if (i < N) dinv[i] = rsqrtf((float)deg[i]);   // deg >= 1 always
}

// ------------------------------------------------------- WMMA GEMM: y=xW*d ----
// One wave computes one 16-row tile across all 64 output channels using
// V_WMMA_F32_16X16X4_F32 (full f32 precision). W is staged in LDS pre-swizzled
// so each lane's B fragment (2 consecutive K values for its N column) is a
// contiguous 8-byte pair -> single ds_load_b64, no packing moves.
// Swizzled layout: pair index p = (kk*4 + c)*32 + lane holds
//   { W[4kk + 2*(lane>>4) + 0][16c + (lane&15)],
//     W[4kk + 2*(lane>>4) + 1][16c + (lane&15)] }
__global__ void __launch_bounds__(256) gemm_xw_kernel(const float* __restrict__ x,
                                                      const float* __restrict__ W,
                                                      const float* __restrict__ dinv,
                                                      float* __restrict__ y,
                                                      float* __restrict__ acc,
                                                      int nRowTiles, int N) {
  __shared__ float Wl[FD * FD];                 // 16 KB, swizzled B fragments
  for (int p = threadIdx.x; p < 2048; p += 256) {
    const int lp   = p & 31;                    // lane this pair belongs to
    const int f    = p >> 5;                    // fragment id = kk*4 + c
    const int c    = f & 3;
    const int kk   = f >> 2;
    const int n    = lp & 15;
    const int h    = lp >> 4;
    const int row  = (kk << 2) + (h << 1);
    const int col  = (c << 4) + n;
    Wl[2 * p + 0] = W[(row + 0) * FD + col];
    Wl[2 * p + 1] = W[(row + 1) * FD + col];
  }
  __syncthreads();

  const int wave = threadIdx.x >> 5;
  const int lane = threadIdx.x & 31;
  const int tile = blockIdx.x * WPB + wave;
  if (tile >= nRowTiles) return;                // wave-uniform: EXEC all-1s below

  const int r0   = tile << 4;
  const int m    = lane & 15;                   // A-row / C-col index within tile
  const int half = lane >> 4;                   // 0: K=0,1 | 1: K=2,3 (f32 layouts)
  const int kb   = half << 1;

  int arow = r0 + m; if (arow >= N) arow = N - 1;   // clamp (N % 16 == 0 anyway)
  const float* __restrict__ xrow  = x + (size_t)arow * FD;
  const float* __restrict__ bbase = Wl + (lane << 1);

  v8f c0 = {}, c1 = {}, c2 = {}, c3 = {};
#pragma unroll
  for (int kk = 0; kk < 16; ++kk) {             // K chunks of 4
    v2f a;                                      // A 16x4: lanes 0-15 K=4kk,4kk+1; 16-31 +2
    a.x = xrow[(kk << 2) + kb + 0];
    a.y = xrow[(kk << 2) + kb + 1];
    const float* fb = bbase + (kk << 8);        // (kk*4)*64 floats
    const v2f b0 = *(const v2f*)(fb + 0 * 64);
    const v2f b1 = *(const v2f*)(fb + 1 * 64);
    const v2f b2 = *(const v2f*)(fb + 2 * 64);
    const v2f b3 = *(const v2f*)(fb + 3 * 64);
    c0 = __builtin_amdgcn_wmma_f32_16x16x4_f32(false, a, false, b0, (short)0, c0, false, false);
    c1 = __builtin_amdgcn_wmma_f32_16x16x4_f32(false, a, false, b1, (short)0, c1, false, false);
    c2 = __builtin_amdgcn_wmma_f32_16x16x4_f32(false, a, false, b2, (short)0, c2, false, false);
    c3 = __builtin_amdgcn_wmma_f32_16x16x4_f32(false, a, false, b3, (short)0, c3, false, false);
  }

#pragma unroll
  for (int r = 0; r < 8; ++r) {                 // C/D: VGPR r -> rows r (lanes 0-15), r+8 (16-31)
    const int row = r0 + r + (half << 3);
    if (row < N) {
      const float s = dinv[row];
      const size_t base = (size_t)row * FD + m;
      float v;
      v = c0[r] * s; y[base +  0] = v; acc[base +  0] = v;
      v = c1[r] * s; y[base + 16] = v; acc[base + 16] = v;
      v = c2[r] * s; y[base + 32] = v; acc[base + 32] = v;
      v = c3[r] * s; y[base + 48] = v; acc[base + 48] = v;
    }
  }
}

// ------------------------------------------------------------ edge scatter ----
// 16 threads per edge; each handles 4 channels: float4 gather of y[src],
// 4 hardware f32 atomics into acc[dst] (L2-resident: 25.6MB << 192MB).
__global__ void __launch_bounds__(256) scatter_kernel(const int* __restrict__ ei, int E,
                                                      const float* __restrict__ y,
                                                      float* __restrict__ acc) {
  long long t = (long long)blockIdx.x * 256 + threadIdx.x;
  int e = (int)(t >> 4);
  if (e >= E) return;
  const int c4 = (int)(t & 15) << 2;
  const int s  = ei[e];
  const int d  = ei[E + e];
  const float4 v = *(const float4*)(y + (size_t)s * FD + c4);
  float* p = acc + (size_t)d * FD + c4;
  unsafeAtomicAdd(p + 0, v.x);
  unsafeAtomicAdd(p + 1, v.y);
  unsafeAtomicAdd(p + 2, v.z);
  unsafeAtomicAdd(p + 3, v.w);
}

// ---------------------------------------------------------------- finalize ----
__global__ void __launch_bounds__(256) finalize_kernel(const float* __restrict__ acc,
                                                       const float* __restrict__ dinv,
                                                       const float* __restrict__ b,
                                                       const float* __restrict__ prelu_a,
                                                       float* __restrict__ out, int N) {
  long long t = (long long)blockIdx.x * 256 + threadIdx.x;
  if (t >= (long long)N * FD) return;
  const int n = (int)(t >> 6);
  const int c = (int)(t & 63);
  const float v = dinv[n] * acc[t] + b[c];
  out[t] = v >= 0.0f ? v : prelu_a[0] * v;
}

// ------------------------------------------------------------------ launch ----
extern "C" void kernel_launch(void* const* d_in, const int* in_sizes, int n_in,
                              void* d_out, int out_size, void* d_ws, size_t ws_size,
                              hipStream_t stream) {
  const float* x  = (const float*)d_in[0];
  const int*   ei = (const int*)d_in[1];     // edge_index [2, E]
  const float* W  = (const float*)d_in[2];
  const float* b  = (const float*)d_in[3];
  const float* pa = (const float*)d_in[4];
  float* out = (float*)d_out;

  const int N = in_sizes[0] / FD;            // 100000
  const int E = in_sizes[1] / 2;             // 1600000

  // Workspace layout: deg[N] u32 | dinv[N] f32 | y[N*64] f32 | acc[N*64] f32  (~52 MB)
  unsigned* deg  = (unsigned*)d_ws;
  float*    dinv = (float*)d_ws + N;
  float*    y    = (float*)d_ws + 2 * (size_t)N;
  float*    acc  = y + (size_t)N * FD;

  const int bN = (N + 255) / 256;
  const int bE = (E + 255) / 256;
  deg_init_kernel<<<bN, 256, 0, stream>>>(deg, N);
  deg_count_kernel<<<bE, 256, 0, stream>>>(ei, E, deg);
  dinv_kernel<<<bN, 256, 0, stream>>>(deg, dinv, N);

  const int nRowTiles = (N + 15) / 16;
  const int bG = (nRowTiles + WPB - 1) / WPB;
  gemm_xw_kernel<<<bG, 256, 0, stream>>>(x, W, dinv, y, acc, nRowTiles, N);

  const long long scatterThreads = (long long)E * 16;
  const int bS = (int)((scatterThreads + 255) / 256);
  scatter_kernel<<<bS, 256, 0, stream>>>(ei, E, y, acc);

  const long long finThreads = (long long)N * FD;
  const int bF = (int)((finThreads + 255) / 256);
  finalize_kernel<<<bF, 256, 0, stream>>>(acc, dinv, b, pa, out, N);
}